// VisionTokenController_32487132627579
// MI455X (gfx1250) — compile-verified
//
#include <hip/hip_runtime.h>
#include <math.h>

#define B_   256
#define N_   576
#define C_   1024
#define NP2_ 578          // N + 2
#define TAU_INV (1.0f / 5.0f)

typedef __attribute__((ext_vector_type(2))) float v2f;
typedef __attribute__((ext_vector_type(4))) float v4f;
typedef __attribute__((ext_vector_type(8))) float v8f;

// ---------------------------------------------------------------------------
// Kernel 1: logits[256,576] = budget_repr[256,1024] @ W[576,1024]^T + b[576]
// One wave (32 lanes) per 16x16 output tile, fp32 WMMA 16x16x4, K-loop = 256.
// A 16x4 layout: lanes 0-15 M=0..15 (K = k+0, k+1), lanes 16-31 (K = k+2, k+3)
// B 4x16 layout mirrored: lane holds N = lane&15, same K split.
// C/D: VGPR r -> M = r + 8*(lane>=16), N = lane&15.
// ---------------------------------------------------------------------------
__global__ __launch_bounds__(32) void logits_wmma_kernel(
    const float* __restrict__ vis, const float* __restrict__ W,
    const float* __restrict__ bias, float* __restrict__ logits) {
  const int tileM = blockIdx.x;          // 0..15  (256 / 16)
  const int tileN = blockIdx.y;          // 0..35  (576 / 16)
  const int lane  = threadIdx.x;         // 0..31
  const int mn    = lane & 15;
  const int kOff  = (lane >> 4) * 2;     // 0 or 2

  // budget_repr row for this lane's M (last token of vision_output)
  const float* Arow = vis + ((size_t)(tileM * 16 + mn) * NP2_ + (NP2_ - 1)) * C_;
  // W row for this lane's N (B = W^T, so B[k][n] = W[n][k])
  const float* Brow = W + (size_t)(tileN * 16 + mn) * C_;

  v8f acc = {};
  for (int k = 0; k < C_; k += 4) {
    v2f a = *(const v2f*)(Arow + k + kOff);   // A[M][k+kOff .. +1]
    v2f b = *(const v2f*)(Brow + k + kOff);   // B[k+kOff .. +1][N]
    acc = __builtin_amdgcn_wmma_f32_16x16x4_f32(
        /*neg_a=*/false, a, /*neg_b=*/false, b,
        /*c_mod=*/(short)0, acc, /*reuse_a=*/false, /*reuse_b=*/false);
  }

  const int nIdx  = tileN * 16 + mn;
  const int mBase = tileM * 16 + ((lane >> 4) * 8);
  const float bb  = bias[nIdx];
#pragma unroll
  for (int r = 0; r < 8; ++r) {
    logits[(size_t)(mBase + r) * N_ + nIdx] = acc[r] + bb;
  }
}

// ---------------------------------------------------------------------------
// Kernel 2: per-row top-K hard mask + softmax -> keep_mask[256,576]
// One block of 576 threads per batch row. Stable rank = #{L[i]>L[j]} +
// #{i<j : L[i]==L[j]} matches jnp double-argsort semantics.
// ---------------------------------------------------------------------------
__global__ __launch_bounds__(N_) void mask_kernel(
    const float* __restrict__ logits, const float* __restrict__ budget,
    float* __restrict__ keep) {
  __shared__ float L[N_];
  __shared__ float E[N_];
  const int b = blockIdx.x;
  const int j = threadIdx.x;

  const float lj = logits[(size_t)b * N_ + j];
  L[j] = lj;
  __syncthreads();

  int cnt = 0;
  float mx = -INFINITY;
  for (int i = 0; i < N_; ++i) {
    const float li = L[i];
    cnt += (li > lj) || (li == lj && i < j);
    mx = fmaxf(mx, li);
  }

  int K = (int)(budget[b] * (float)N_);   // trunc toward zero, like astype(int32)
  K = min(max(K, 1), N_);
  const float hard = (cnt < K) ? 1.0f : 0.0f;

  const float e = expf((lj - mx) * TAU_INV);
  E[j] = e;
  __syncthreads();
  for (int s = 512; s > 0; s >>= 1) {
    if (j < s && (j + s) < N_) E[j] += E[j + s];
    __syncthreads();
  }
  const float soft = e / E[0];

  // keep the reference's fp expression shape: (hard + soft) - soft
  keep[(size_t)b * N_ + j] = (hard + soft) - soft;
}

// ---------------------------------------------------------------------------
// Kernel 3: masked_patches[b,j,:] = vision_output[b,j+1,:] * keep[b,j]
// One 256-thread block per (b,j) row: 256 lanes x float4 = 1024 floats.
// Nontemporal b128 stream (1.2 GB total) — the bandwidth-bound part.
// ---------------------------------------------------------------------------
__global__ __launch_bounds__(256) void scale_kernel(
    const float* __restrict__ vis, const float* __restrict__ keep,
    float* __restrict__ out) {
  const int row = blockIdx.x;            // 0 .. B*N-1 (uniform -> scalar loads)
  const int b   = row / N_;
  const int j   = row - b * N_;
  const float m = keep[row];

  const v4f* src = (const v4f*)(vis + ((size_t)b * NP2_ + (size_t)(j + 1)) * C_);
  v4f*       dst = (v4f*)(out + (size_t)row * C_);

  v4f v = __builtin_nontemporal_load(src + threadIdx.x);
  v.x *= m; v.y *= m; v.z *= m; v.w *= m;
  __builtin_nontemporal_store(v, dst + threadIdx.x);
}

// ---------------------------------------------------------------------------
extern "C" void kernel_launch(void* const* d_in, const int* in_sizes, int n_in,
                              void* d_out, int out_size, void* d_ws, size_t ws_size,
                              hipStream_t stream) {
  const float* vis    = (const float*)d_in[0];  // (256, 578, 1024)
  const float* budget = (const float*)d_in[1];  // (256,)
  const float* W      = (const float*)d_in[2];  // (576, 1024)
  const float* bias   = (const float*)d_in[3];  // (576,)

  float* out  = (float*)d_out;                          // masked_patches
  float* keep = out + (size_t)B_ * N_ * C_;             // keep_mask tail
  float* logits = (float*)d_ws;                         // 256*576 floats scratch

  logits_wmma_kernel<<<dim3(B_ / 16, N_ / 16), 32, 0, stream>>>(vis, W, bias, logits);
  mask_kernel<<<B_, N_, 0, stream>>>(logits, budget, keep);
  scale_kernel<<<B_ * N_, 256, 0, stream>>>(vis, keep, out);
}